// GRU2d_84902913507522
// MI455X (gfx1250) — compile-verified
//
#include <hip/hip_runtime.h>
#include <math.h>

// ---------------- problem constants ----------------
#define Bsz   2
#define CIN   64
#define COUT  64
#define Hd    128
#define Wd    128
#define HW    (Hd*Wd)      // 16384
#define Nn    8            // directions
#define HSE   16           // C_OUT/FACT
#define HID   128          // HID_ATTN
#define GATT  16           // G_ATTN

typedef __attribute__((ext_vector_type(2))) float v2f;
typedef __attribute__((ext_vector_type(8))) float v8f;

__device__ __forceinline__ float sigm(float v) { return 1.0f / (1.0f + __expf(-v)); }

// =====================================================================
// K1: y_h = h_en_w @ x ; y_z = z_en_w @ x  (1x1 conv as GEMM, fp32 WMMA)
// Weight-stationary tiling: each wave owns one 16-out-channel tile (mt),
// preloads its A-fragments for all 16 k-steps (both weight matrices,
// 64 VGPRs) once, then sweeps 8 pixel tiles of 16: load B, 2x WMMA, store.
// A (16x4): lane m=L%16, k = 2*(L/16)+j.  B (4x16): lane n=L%16, k = 2*(L/16)+j.
// C/D (16x16): vgpr r -> m = r + 8*(L/16), n = L%16.
// =====================================================================
#define TILES_PER_WAVE 8
__global__ void __launch_bounds__(256) conv_en_wmma(
    const float* __restrict__ x, const float* __restrict__ wh,
    const float* __restrict__ wz, float* __restrict__ yh, float* __restrict__ yz)
{
  int gw   = (blockIdx.x * 256 + (int)threadIdx.x) >> 5;   // global wave id
  int lane = threadIdx.x & 31;
  // gw = (((b*4) + mt) * 128 + chunk)
  int b     = gw >> 9;           // 512 waves per batch
  int rem   = gw & 511;
  int mt    = rem >> 7;          // 4 out-channel tiles
  int chunk = rem & 127;         // 128 chunks of 8 pixel-tiles
  int t0    = chunk * TILES_PER_WAVE;

  int kh = (lane >> 4) << 1;     // 0 or 2
  int m  = lane & 15;
  int mm = mt * 16 + m;
  const float* xb = x + (size_t)b * CIN * HW;

  // preload A fragments (weights) for all 16 k-steps, both matrices
  v2f ahf[16], azf[16];
  #pragma unroll
  for (int ks = 0; ks < 16; ++ks) {
    ahf[ks][0] = wh[mm * CIN + ks * 4 + kh];
    ahf[ks][1] = wh[mm * CIN + ks * 4 + kh + 1];
    azf[ks][0] = wz[mm * CIN + ks * 4 + kh];
    azf[ks][1] = wz[mm * CIN + ks * 4 + kh + 1];
  }

  int ob = mt * 16 + ((lane >> 4) << 3);   // first of 8 output rows this lane holds
  for (int tt = 0; tt < TILES_PER_WAVE; ++tt) {
    int n = (t0 + tt) * 16 + (lane & 15);
    v8f ch = {}; v8f cz = {};
    #pragma unroll
    for (int ks = 0; ks < 16; ++ks) {
      v2f bt;
      bt[0] = xb[(ks * 4 + kh + 0) * HW + n];
      bt[1] = xb[(ks * 4 + kh + 1) * HW + n];
      ch = __builtin_amdgcn_wmma_f32_16x16x4_f32(false, ahf[ks], false, bt,
                                                 (short)0, ch, false, false);
      cz = __builtin_amdgcn_wmma_f32_16x16x4_f32(false, azf[ks], false, bt,
                                                 (short)0, cz, false, false);
    }
    #pragma unroll
    for (int r = 0; r < 8; ++r) {
      yh[((size_t)b * COUT + ob + r) * HW + n] = ch[r];
      yz[((size_t)b * COUT + ob + r) * HW + n] = cz[r];
    }
  }
}

// =====================================================================
// K2: per-(b,c) mean & max over H*W for both y_h and y_z
// stats layout: [4][B*64] = mean_h, max_h, mean_z, max_z
// =====================================================================
__global__ void __launch_bounds__(256) pool_kernel(
    const float* __restrict__ yh, const float* __restrict__ yz, float* __restrict__ stats)
{
  int bc = blockIdx.x, t = threadIdx.x;
  const float* A = yh + (size_t)bc * HW;
  const float* Z = yz + (size_t)bc * HW;
  float sh = 0.f, mh = -3.4e38f, sz = 0.f, mz = -3.4e38f;
  for (int p = t; p < HW; p += 256) {
    float a = A[p]; sh += a; mh = fmaxf(mh, a);
    float v = Z[p]; sz += v; mz = fmaxf(mz, v);
  }
  __shared__ float r0[256], r1[256], r2[256], r3[256];
  r0[t] = sh; r1[t] = mh; r2[t] = sz; r3[t] = mz;
  __syncthreads();
  for (int o = 128; o > 0; o >>= 1) {
    if (t < o) {
      r0[t] += r0[t + o];              r1[t] = fmaxf(r1[t], r1[t + o]);
      r2[t] += r2[t + o];              r3[t] = fmaxf(r3[t], r3[t + o]);
    }
    __syncthreads();
  }
  if (t == 0) {
    stats[0 * (Bsz*COUT) + bc] = r0[0] / (float)HW;
    stats[1 * (Bsz*COUT) + bc] = r1[0];
    stats[2 * (Bsz*COUT) + bc] = r2[0] / (float)HW;
    stats[3 * (Bsz*COUT) + bc] = r3[0];
  }
}

// =====================================================================
// K3: SE squeeze/excite MLPs (both paths): hdn = relu(s_w @ [mean;max]);
// attn = sigmoid(gconv(e_w, hdn, groups=8)) -> attn[b][n*64+c]
// =====================================================================
__global__ void __launch_bounds__(128) attn_kernel(
    const float* __restrict__ stats,
    const float* __restrict__ swh, const float* __restrict__ ewh,
    const float* __restrict__ swz, const float* __restrict__ ewz,
    float* __restrict__ attn_h, float* __restrict__ attn_z)
{
  int b = blockIdx.x, t = threadIdx.x;   // t = hidden unit 0..127
  __shared__ float hdnh[HID], hdnz[HID];
  float ah = 0.f, az = 0.f;
  for (int c = 0; c < 2 * COUT; ++c) {
    float ph = (c < COUT) ? stats[0 * (Bsz*COUT) + b * COUT + c]
                          : stats[1 * (Bsz*COUT) + b * COUT + (c - COUT)];
    float pz = (c < COUT) ? stats[2 * (Bsz*COUT) + b * COUT + c]
                          : stats[3 * (Bsz*COUT) + b * COUT + (c - COUT)];
    ah += swh[t * (2*COUT) + c] * ph;
    az += swz[t * (2*COUT) + c] * pz;
  }
  hdnh[t] = fmaxf(ah, 0.f);
  hdnz[t] = fmaxf(az, 0.f);
  __syncthreads();
  #pragma unroll
  for (int j = 0; j < 4; ++j) {
    int oc = t + j * 128;                // 0..511 == n*64+c
    int g  = oc >> 6;                    // group n
    float sh = 0.f, sz = 0.f;
    #pragma unroll
    for (int i = 0; i < HSE; ++i) {
      sh += ewh[oc * HSE + i] * hdnh[g * HSE + i];
      sz += ewz[oc * HSE + i] * hdnz[g * HSE + i];
    }
    attn_h[b * (Nn*COUT) + oc] = sigm(sh);
    attn_z[b * (Nn*COUT) + oc] = sigm(sz);
  }
}

// =====================================================================
// K4a: axis GRU scan. axis=0: scan over y (thread=x); axis=1: scan over x
// (thread=y). neg: reversed scan. Fuses hh = y_h*attn_h, z = sigm(y_z*attn_z).
// =====================================================================
__global__ void __launch_bounds__(128) gru_axis(
    const float* __restrict__ yh, const float* __restrict__ yz,
    const float* __restrict__ attn_h, const float* __restrict__ attn_z,
    const float* __restrict__ h0, float* __restrict__ xcat,
    int i, int axis, int neg)
{
  int bc = blockIdx.x;                 // b*64 + c
  int b = bc >> 6, c = bc & 63;
  int lane = threadIdx.x;              // fixed coordinate
  float ah = attn_h[b * (Nn*COUT) + i * COUT + c];
  float az = attn_z[b * (Nn*COUT) + i * COUT + c];
  float h  = h0[i * COUT + c];
  const float* Yh = yh + (size_t)bc * HW;
  const float* Yz = yz + (size_t)bc * HW;
  float* O = xcat + ((size_t)b * (Nn*COUT) + i * COUT + c) * HW;
  for (int s = 0; s < 128; ++s) {
    int t = neg ? (127 - s) : s;
    int p = (axis == 0) ? (t * Wd + lane) : (lane * Wd + t);
    float z  = sigm(Yz[p] * az);
    float hh = Yh[p] * ah;
    h = z * hh + (1.0f - z) * h;
    O[p] = h;
  }
}

// =====================================================================
// K4b: diagonal GRU scan. Block per (b,c), 128 threads over x, LDS prev row.
// h[y][x] depends on h[y-d1][x-d2]; boundaries get h0.
// =====================================================================
__global__ void __launch_bounds__(128) gru_diag(
    const float* __restrict__ yh, const float* __restrict__ yz,
    const float* __restrict__ attn_h, const float* __restrict__ attn_z,
    const float* __restrict__ h0, float* __restrict__ xcat,
    int i, int d1, int d2)
{
  __shared__ float prev[Wd];
  int bc = blockIdx.x;
  int b = bc >> 6, c = bc & 63;
  int xx = threadIdx.x;
  float ah  = attn_h[b * (Nn*COUT) + i * COUT + c];
  float az  = attn_z[b * (Nn*COUT) + i * COUT + c];
  float h0v = h0[i * COUT + c];
  const float* Yh = yh + (size_t)bc * HW;
  const float* Yz = yz + (size_t)bc * HW;
  float* O = xcat + ((size_t)b * (Nn*COUT) + i * COUT + c) * HW;
  prev[xx] = h0v;
  __syncthreads();
  for (int s = 0; s < 128; ++s) {
    int y  = (d1 > 0) ? s : (127 - s);
    int xn = (d2 > 0) ? (xx - 1) : (xx + 1);
    float hp = (xn < 0 || xn > 127) ? h0v : prev[xn];
    int p = y * Wd + xx;
    float z  = sigm(Yz[p] * az);
    float hh = Yh[p] * ah;
    float hn = z * hh + (1.0f - z) * hp;
    __syncthreads();           // everyone done reading prev
    prev[xx] = hn;
    __syncthreads();           // new row visible
    O[p] = hn;
  }
}

// =====================================================================
// K5: per-pixel channel sort (512, bitonic, thread-private LDS row) +
// grouped-conv attention chain (shuffle folded into read index) + output.
// Row layout: [0..511] sorted copy, [512..639] bufA, [640..767] bufB, pad->769
// (stride 769 ≡ 1 mod 64 -> conflict-free per-lane rows).
// =====================================================================
#define ROWSTRIDE 769
__global__ void __launch_bounds__(64) attn_sort_out(
    const float* __restrict__ xcat,
    const float* __restrict__ w1, const float* __restrict__ w2,
    const float* __restrict__ w3, const float* __restrict__ w4,
    float* __restrict__ out)
{
  extern __shared__ float lds[];
  int t = threadIdx.x;
  int b = blockIdx.x >> 8;                      // 256 blocks per batch
  int p = ((blockIdx.x & 255) << 6) + t;        // pixel
  const float* X = xcat + (size_t)b * (Nn*COUT) * HW;
  float* row  = lds + t * ROWSTRIDE;
  float* bufA = row + 512;
  float* bufB = bufA + 128;

  for (int ch = 0; ch < 512; ++ch) row[ch] = X[(size_t)ch * HW + p]; // coalesced

  // serial bitonic sort ascending over own row
  for (int k = 2; k <= 512; k <<= 1)
    for (int j = k >> 1; j > 0; j >>= 1)
      for (int idx = 0; idx < 512; ++idx) {
        int l = idx ^ j;
        if (l > idx) {
          float a = row[idx], v = row[l];
          bool up = ((idx & k) == 0);
          if (up ? (a > v) : (a < v)) { row[idx] = v; row[l] = a; }
        }
      }

  // stage1: t=[xcat;sorted](1024ch) -> gconv groups=32 (in 32, out 4) -> bufA[128]
  for (int g = 0; g < 32; ++g) {
    float a0 = 0.f, a1 = 0.f, a2 = 0.f, a3 = 0.f;
    #pragma unroll
    for (int ii = 0; ii < 32; ++ii) {
      float v = (g < 16) ? X[(size_t)(g * 32 + ii) * HW + p]   // xcat half (L2 hit)
                         : row[(g - 16) * 32 + ii];            // sorted half
      a0 += w1[(g * 4 + 0) * 32 + ii] * v;
      a1 += w1[(g * 4 + 1) * 32 + ii] * v;
      a2 += w1[(g * 4 + 2) * 32 + ii] * v;
      a3 += w1[(g * 4 + 3) * 32 + ii] * v;
    }
    bufA[g * 4 + 0] = a0; bufA[g * 4 + 1] = a1;
    bufA[g * 4 + 2] = a2; bufA[g * 4 + 3] = a3;
  }
  // stage2: shuffle(16) then gconv groups=16 (in 8, out 8): bufB = w2 * shuf(bufA)
  for (int o = 0; o < 128; ++o) {
    int g = o >> 3;
    float acc = 0.f;
    #pragma unroll
    for (int ii = 0; ii < 8; ++ii) {
      int cs = g * 8 + ii;                       // shuffled index
      acc += w2[o * 8 + ii] * bufA[(cs & 15) * 8 + (cs >> 4)];
    }
    bufB[o] = acc;
  }
  // stage3: relu(w3 * shuf(bufB)) -> bufA
  for (int o = 0; o < 128; ++o) {
    int g = o >> 3;
    float acc = 0.f;
    #pragma unroll
    for (int ii = 0; ii < 8; ++ii) {
      int cs = g * 8 + ii;
      acc += w3[o * 8 + ii] * bufB[(cs & 15) * 8 + (cs >> 4)];
    }
    bufA[o] = fmaxf(acc, 0.f);
  }
  // stage4: scores = sigmoid(w4 * shuf(bufA)); y[c] = sum_n xcat[n*64+c]*score
  for (int c = 0; c < COUT; ++c) {
    float a = 0.f;
    #pragma unroll
    for (int n = 0; n < Nn; ++n) {
      int s = n * COUT + c;
      int g = s >> 5;
      float sc = 0.f;
      #pragma unroll
      for (int ii = 0; ii < 8; ++ii) {
        int cs = g * 8 + ii;
        sc += w4[s * 8 + ii] * bufA[(cs & 15) * 8 + (cs >> 4)];
      }
      a += X[(size_t)s * HW + p] * sigm(sc);
    }
    out[((size_t)b * COUT + c) * HW + p] = a;
  }
}

// =====================================================================
extern "C" void kernel_launch(void* const* d_in, const int* in_sizes, int n_in,
                              void* d_out, int out_size, void* d_ws, size_t ws_size,
                              hipStream_t stream) {
  (void)in_sizes; (void)n_in; (void)out_size;
  const float* x    = (const float*)d_in[0];
  const float* h0   = (const float*)d_in[1];
  const float* h_en = (const float*)d_in[2];
  const float* h_s  = (const float*)d_in[3];
  const float* h_e  = (const float*)d_in[4];
  const float* z_en = (const float*)d_in[5];
  const float* z_s  = (const float*)d_in[6];
  const float* z_e  = (const float*)d_in[7];
  const float* w1   = (const float*)d_in[8];
  const float* w2   = (const float*)d_in[9];
  const float* w3   = (const float*)d_in[10];
  const float* w4   = (const float*)d_in[11];
  float* out = (float*)d_out;

  // workspace carving (floats)
  float* yh    = (float*)d_ws;                          // B*64*HW
  float* yz    = yh + (size_t)Bsz * COUT * HW;          // B*64*HW
  float* stats = yz + (size_t)Bsz * COUT * HW;          // 4*B*64
  float* ath   = stats + 4 * Bsz * COUT;                // B*512
  float* atz   = ath + (size_t)Bsz * Nn * COUT;         // B*512
  float* xcat  = atz + (size_t)Bsz * Nn * COUT;         // B*512*HW
  size_t need  = (size_t)(xcat - yh + (size_t)Bsz * Nn * COUT * HW) * sizeof(float);
  if (ws_size < need) return;

  // waves = B * 4(mt) * 128(chunks) = 1024 -> 128 blocks of 8 waves
  conv_en_wmma<<<128, 256, 0, stream>>>(x, h_en, z_en, yh, yz);
  pool_kernel<<<Bsz * COUT, 256, 0, stream>>>(yh, yz, stats);
  attn_kernel<<<Bsz, 128, 0, stream>>>(stats, h_s, h_e, z_s, z_e, ath, atz);

  // DIRECTIONS = [(1,1),(1,-1),(-1,1),(-1,-1),(1,0),(-1,0),(0,1),(0,-1)]
  gru_diag<<<Bsz * COUT, 128, 0, stream>>>(yh, yz, ath, atz, h0, xcat, 0,  1,  1);
  gru_diag<<<Bsz * COUT, 128, 0, stream>>>(yh, yz, ath, atz, h0, xcat, 1,  1, -1);
  gru_diag<<<Bsz * COUT, 128, 0, stream>>>(yh, yz, ath, atz, h0, xcat, 2, -1,  1);
  gru_diag<<<Bsz * COUT, 128, 0, stream>>>(yh, yz, ath, atz, h0, xcat, 3, -1, -1);
  gru_axis<<<Bsz * COUT, 128, 0, stream>>>(yh, yz, ath, atz, h0, xcat, 4, 0, 0); // (1,0)
  gru_axis<<<Bsz * COUT, 128, 0, stream>>>(yh, yz, ath, atz, h0, xcat, 5, 0, 1); // (-1,0)
  gru_axis<<<Bsz * COUT, 128, 0, stream>>>(yh, yz, ath, atz, h0, xcat, 6, 1, 0); // (0,1)
  gru_axis<<<Bsz * COUT, 128, 0, stream>>>(yh, yz, ath, atz, h0, xcat, 7, 1, 1); // (0,-1)

  attn_sort_out<<<Bsz * 256, 64, 64 * ROWSTRIDE * sizeof(float), stream>>>(
      xcat, w1, w2, w3, w4, out);
}